// FIPool_66108136620095
// MI455X (gfx1250) — compile-verified
//
#include <hip/hip_runtime.h>
#include <math.h>

typedef float v2f __attribute__((ext_vector_type(2)));
typedef float v8f __attribute__((ext_vector_type(8)));

#define BB 8
#define NN 1024
#define CC 256
#define KMAX 256
#define TWO_C 512

// ---------------------------------------------------------------------------
// Kernel 1: xsum16[b][c] = sum_n X[b,n,c]  (rows 8..15 zero-padded for WMMA M=16)
// grid 16 x 256 threads, coalesced over c.
// ---------------------------------------------------------------------------
__global__ void k_xsum(const float* __restrict__ X, float* __restrict__ xsum16) {
  int b = blockIdx.x;   // 0..15
  int c = threadIdx.x;  // 0..255
  float s = 0.f;
  if (b < BB) {
    const float* xb = X + (size_t)b * NN * CC + c;
#pragma unroll 4
    for (int n = 0; n < NN; ++n) s += xb[(size_t)n * CC];
  }
  xsum16[b * CC + c] = s;
}

// ---------------------------------------------------------------------------
// Kernel 2a: ksum16 = xsum16 (16xC) @ Wk (CxC), Wk = Wqkv[:, C:2C]
// One wave (32 threads) per 16-column tile; V_WMMA_F32_16X16X4_F32, K stepped by 4.
// A frag: lanes 0-15 -> M=lane, K=k0+{0,1}; lanes 16-31 -> M=lane-16, K=k0+{2,3}
// B frag: lanes 0-15 -> K=k0+{0,1}(vgpr0/1), N=lane; lanes 16-31 -> K=k0+{2,3}, N=lane-16
// D: vgpr v -> M = v + 8*(lane>=16), N = lane&15
// ---------------------------------------------------------------------------
__global__ void k_gemm_ksum(const float* __restrict__ A16,
                            const float* __restrict__ Wqkv,
                            float* __restrict__ D16) {
  int nt   = blockIdx.x;        // 0..15 column tile
  int lane = threadIdx.x;       // 0..31
  int nn   = lane & 15;
  int kh   = (lane >> 4) * 2;   // 0 or 2
  int c0   = nt * 16;
  v8f acc = {};
  for (int k0 = 0; k0 < CC; k0 += 4) {
    v2f a, b;
    a.x = A16[nn * CC + k0 + kh];
    a.y = A16[nn * CC + k0 + kh + 1];
    b.x = Wqkv[(size_t)(k0 + kh)     * TWO_C + CC + c0 + nn];
    b.y = Wqkv[(size_t)(k0 + kh + 1) * TWO_C + CC + c0 + nn];
    acc = __builtin_amdgcn_wmma_f32_16x16x4_f32(false, a, false, b,
                                                (short)0, acc, false, false);
  }
  int mbase = (lane >> 4) * 8;
#pragma unroll
  for (int v = 0; v < 8; ++v)
    D16[(mbase + v) * CC + c0 + nn] = acc[v];
}

// ---------------------------------------------------------------------------
// Kernel 2b: w16 = ksum16 (16xC) @ Wq^T, i.e. B[k][n] = Wqkv[c0+n][k]
// ---------------------------------------------------------------------------
__global__ void k_gemm_w(const float* __restrict__ A16,
                         const float* __restrict__ Wqkv,
                         float* __restrict__ D16) {
  int nt   = blockIdx.x;
  int lane = threadIdx.x;
  int nn   = lane & 15;
  int kh   = (lane >> 4) * 2;
  int c0   = nt * 16;
  v8f acc = {};
  for (int k0 = 0; k0 < CC; k0 += 4) {
    v2f a, b;
    a.x = A16[nn * CC + k0 + kh];
    a.y = A16[nn * CC + k0 + kh + 1];
    b.x = Wqkv[(size_t)(c0 + nn) * TWO_C + (k0 + kh)];
    b.y = Wqkv[(size_t)(c0 + nn) * TWO_C + (k0 + kh + 1)];
    acc = __builtin_amdgcn_wmma_f32_16x16x4_f32(false, a, false, b,
                                                (short)0, acc, false, false);
  }
  int mbase = (lane >> 4) * 8;
#pragma unroll
  for (int v = 0; v < 8; ++v)
    D16[(mbase + v) * CC + c0 + nn] = acc[v];
}

// ---------------------------------------------------------------------------
// Kernel 3: scores[b,n] = mask ? scale * dot(X[b,n,:], w16[b,:]) : -1e9
// One wave32 per row; butterfly reduce.
// ---------------------------------------------------------------------------
__global__ void k_scores(const float* __restrict__ X,
                         const float* __restrict__ mask,
                         const float* __restrict__ w16,
                         float* __restrict__ scores) {
  int lane = threadIdx.x & 31;
  int wave = threadIdx.x >> 5;
  int row  = blockIdx.x * 8 + wave;   // 0..8191
  int b = row >> 10;
  int n = row & 1023;
  const float* x = X + (size_t)row * CC;
  const float* w = w16 + b * CC;
  float s = 0.f;
#pragma unroll
  for (int c = lane; c < CC; c += 32) s += x[c] * w[c];
#pragma unroll
  for (int off = 16; off >= 1; off >>= 1) s += __shfl_xor(s, off, 32);
  if (lane == 0) {
    const float scale = 0.17677669529663687f;  // (256/8)^-0.5
    scores[row] = (mask[b * NN + n] > 0.f) ? (scale * s) : -1.0e9f;
  }
}

// ---------------------------------------------------------------------------
// Kernel 4: exact top-k via rank counting (matches jax.lax.top_k: descending,
// stable lowest-index tie-break). One block of 1024 threads per batch.
// ---------------------------------------------------------------------------
__global__ __launch_bounds__(1024) void k_topk(
    const float* __restrict__ scores, const float* __restrict__ mask,
    int* __restrict__ idx, float* __restrict__ gate, float* __restrict__ nm,
    float* __restrict__ newmask_out) {
  __shared__ float s[NN];
  __shared__ float red[NN];
  int b = blockIdx.x;
  int i = threadIdx.x;
  float si = scores[b * NN + i];
  s[i]   = si;
  red[i] = mask[b * NN + i];
  __syncthreads();
  for (int off = 512; off >= 1; off >>= 1) {
    if (i < off) red[i] += red[i + off];
    __syncthreads();
  }
  int k_i = (int)ceilf(0.25f * red[0]);
  int rank = 0;
  for (int j = 0; j < NN; ++j) {
    float sj = s[j];
    rank += (sj > si) || (sj == si && j < i);
  }
  if (rank < KMAX) {
    int pos = b * KMAX + rank;
    float m = (rank < k_i) ? 1.f : 0.f;
    idx[pos]         = i;
    nm[pos]          = m;
    gate[pos]        = tanhf(si) * m;
    newmask_out[pos] = m;
  }
}

// ---------------------------------------------------------------------------
// Kernel 5: new_X[b,r,c] = X[b, idx[b,r], c] * gate[b,r]   (coalesced over c)
// ---------------------------------------------------------------------------
__global__ void k_gather_x(const float* __restrict__ X, const int* __restrict__ idx,
                           const float* __restrict__ gate, float* __restrict__ outX) {
  int br = blockIdx.x;  // b*KMAX + r
  int c  = threadIdx.x;
  int b  = br >> 8;
  int i  = idx[br];
  float g = gate[br];
  outX[(size_t)br * CC + c] = X[((size_t)b * NN + i) * CC + c] * g;
}

// ---------------------------------------------------------------------------
// Kernel 6: new_adj[b,r,s] = adj[b, idx[r], idx[s]] * nm[r] * nm[s]
// Only ~2 MB of the 32 MB adj is ever touched.
// ---------------------------------------------------------------------------
__global__ void k_gather_adj(const float* __restrict__ adj, const int* __restrict__ idx,
                             const float* __restrict__ nm, float* __restrict__ outA) {
  int br = blockIdx.x;
  int sI = threadIdx.x;  // 0..255
  int b  = br >> 8;
  int i  = idx[br];
  float nr = nm[br];
  int j    = idx[b * KMAX + sI];
  float ns = nm[b * KMAX + sI];
  outA[(size_t)br * KMAX + sI] = adj[((size_t)b * NN + i) * NN + j] * nr * ns;
}

// ---------------------------------------------------------------------------
extern "C" void kernel_launch(void* const* d_in, const int* in_sizes, int n_in,
                              void* d_out, int out_size, void* d_ws, size_t ws_size,
                              hipStream_t stream) {
  const float* X    = (const float*)d_in[0];
  const float* adj  = (const float*)d_in[1];
  const float* mask = (const float*)d_in[2];
  const float* Wqkv = (const float*)d_in[3];

  float* out     = (float*)d_out;
  float* newX    = out;                                           // 8*256*256
  float* newAdj  = out + (size_t)BB * KMAX * CC;                  // 8*256*256
  float* newMask = newAdj + (size_t)BB * KMAX * KMAX;             // 8*256

  float* ws      = (float*)d_ws;
  float* xsum16  = ws;               // 16*256
  float* ksum16  = ws + 4096;        // 16*256
  float* w16     = ws + 8192;        // 16*256
  float* scores  = ws + 12288;       // 8*1024
  float* gate    = ws + 20480;       // 8*256
  float* nm      = ws + 22528;       // 8*256
  int*   idx     = (int*)(ws + 24576); // 8*256 ints

  k_xsum      <<<16,        256, 0, stream>>>(X, xsum16);
  k_gemm_ksum <<<16,        32,  0, stream>>>(xsum16, Wqkv, ksum16);
  k_gemm_w    <<<16,        32,  0, stream>>>(ksum16, Wqkv, w16);
  k_scores    <<<BB*NN/8,   256, 0, stream>>>(X, mask, w16, scores);
  k_topk      <<<BB,        1024,0, stream>>>(scores, mask, idx, gate, nm, newMask);
  k_gather_x  <<<BB*KMAX,   CC,  0, stream>>>(X, idx, gate, newX);
  k_gather_adj<<<BB*KMAX,   KMAX,0, stream>>>(adj, idx, nm, newAdj);
}